// BoxMatcher_12257836663524
// MI455X (gfx1250) — compile-verified
//
#include <hip/hip_runtime.h>
#include <stdint.h>

// Problem constants (from reference setup_inputs)
#define IMG_H 256
#define IMG_W 256
#define BB    8
#define QQ    16384
#define TT    256
#define QPB   64            // queries per block
#define THREADS 256
#define WAVES (THREADS / 32)
#define QPW (QPB / WAVES)   // queries per wave = 8
#define EPSF 1e-7f

// ---- CDNA5 async-to-LDS path (gfx1250): guarded so compile always succeeds ----
#if __has_builtin(__builtin_amdgcn_global_load_async_to_lds_b128) && \
    __has_builtin(__builtin_amdgcn_s_wait_asynccnt)
#define USE_ASYNC_LDS 1
#else
#define USE_ASYNC_LDS 0
#endif

// Builtin wants pointers-to-int4 in explicit address spaces (AS1 global / AS3 LDS)
typedef int   v4i __attribute__((ext_vector_type(4)));
typedef float v4f __attribute__((ext_vector_type(4)));
typedef __attribute__((address_space(1))) v4i GV4;   // global int4
typedef __attribute__((address_space(3))) v4i LV4;   // LDS int4

__global__ __launch_bounds__(THREADS) void
box_matcher_kernel(const float4* __restrict__ qry,   // [B,Q,4]
                   const float4* __restrict__ tgt,   // [B,T,4]
                   const uint8_t* __restrict__ masks,// [B,T,H,W] bool (1B)
                   float* __restrict__ out)          // sim | labels | ids (flat)
{
    __shared__ float4 s_tgt[TT];    // 4 KB: target boxes for this batch
    __shared__ float  s_area[TT];   // 1 KB: target areas
    __shared__ float4 s_qry[QPB];   // 1 KB: this block's query boxes

    const int bid          = blockIdx.x;
    const int blocks_per_b = QQ / QPB;           // 256
    const int b            = bid / blocks_per_b;
    const int q0           = (bid % blocks_per_b) * QPB;
    const int tid          = threadIdx.x;

#if USE_ASYNC_LDS
    // CDNA5 async data path: GLOBAL_LOAD_ASYNC_TO_LDS_B128 (tracked by ASYNCcnt)
    {
        const float4* gt = &tgt[(size_t)b * TT + tid];
        __builtin_amdgcn_global_load_async_to_lds_b128(
            (GV4*)gt, (LV4*)&s_tgt[tid], 0, 0);
        if (tid < QPB) {
            const float4* gq = &qry[(size_t)b * QQ + q0 + tid];
            __builtin_amdgcn_global_load_async_to_lds_b128(
                (GV4*)gq, (LV4*)&s_qry[tid], 0, 0);
        }
        __builtin_amdgcn_s_wait_asynccnt(0);     // s_wait_asynccnt 0
    }
#else
    s_tgt[tid] = tgt[(size_t)b * TT + tid];
    if (tid < QPB) s_qry[tid] = qry[(size_t)b * QQ + q0 + tid];
#endif
    __syncthreads();

    {   // precompute target areas once
        float4 tb = s_tgt[tid];
        s_area[tid] = (tb.z - tb.x) * (tb.w - tb.y);
    }
    __syncthreads();

    const int wave = tid >> 5;
    const int lane = tid & 31;

    float* const sim_base = out;                                 // B*Q*T floats
    float* const labels   = out + (size_t)BB * QQ * TT;          // B*Q floats
    float* const ids      = labels + (size_t)BB * QQ;            // B*Q floats
    const size_t mask_b   = (size_t)b * TT * (size_t)(IMG_H * IMG_W);

    for (int i = 0; i < QPW; ++i) {
        const int ql = wave * QPW + i;      // local query index in block
        const int q  = q0 + ql;             // query index in batch
        const float4 qb = s_qry[ql];
        const float aq  = (qb.z - qb.x) * (qb.w - qb.y);

        // center pixel (trunc toward zero like jnp astype(int32), then clip)
        int cx = (int)((qb.x + qb.z) * 0.5f);
        int cy = (int)((qb.y + qb.w) * 0.5f);
        cx = min(max(cx, 0), IMG_W - 1);
        cy = min(max(cy, 0), IMG_H - 1);
        const int cofs = (cy << 8) | cx;    // cy*W + cx

        float best = -INFINITY;
        int   bidx = 0;
        float* __restrict__ simRow = sim_base + ((size_t)b * QQ + q) * TT;

        // Each lane owns 4 consecutive targets per chunk -> one b128 NT store,
        // 512B per wave per store, fully coalesced. 2 chunks cover T=256.
        #pragma unroll
        for (int c = 0; c < TT / (32 * 4); ++c) {
            const int t0 = c * 128 + lane * 4;
            v4f simv;
            #pragma unroll
            for (int j = 0; j < 4; ++j) {
                const int t = t0 + j;       // ascending t within lane
                const float4 tb = s_tgt[t];
                const float  at = s_area[t];

                float iw = fminf(qb.z, tb.z) - fmaxf(qb.x, tb.x);
                float ih = fminf(qb.w, tb.w) - fmaxf(qb.y, tb.y);
                iw = fmaxf(iw, 0.0f);
                ih = fmaxf(ih, 0.0f);
                const float inter = iw * ih;
                const float uni   = aq + at - inter;
                const float iou   = inter / (uni + EPSF);
                const float ew = fmaxf(qb.z, tb.z) - fminf(qb.x, tb.x);
                const float eh = fmaxf(qb.w, tb.w) - fminf(qb.y, tb.y);
                const float enc = ew * eh;
                const float giou = iou - (enc - uni) / (enc + EPSF);

                // mask gather: planes 64KB apart; array is L2-resident (128MB < 192MB L2)
                const uint8_t m = masks[mask_b + ((size_t)t << 16) + (size_t)cofs];
                const float sim = m ? giou : 0.0f;

                simv[j] = sim;
                if (sim > best) { best = sim; bidx = t; }  // first-occurrence within lane
            }
            // Non-temporal: sim is write-once streaming; keep L2 for mask gathers
            __builtin_nontemporal_store(simv, (v4f*)(simRow + t0));
        }

        // wave32 max/argmax reduction, ties -> smaller t (matches jnp.argmax)
        #pragma unroll
        for (int off = 16; off > 0; off >>= 1) {
            const float ov = __shfl_xor(best, off, 32);
            const int   oi = __shfl_xor(bidx, off, 32);
            if (ov > best || (ov == best && oi < bidx)) { best = ov; bidx = oi; }
        }

        if (lane == 0) {
            const float lab = (best >= 0.5f) ? 1.0f : ((best >= 0.3f) ? -1.0f : 0.0f);
            labels[(size_t)b * QQ + q] = lab;
            ids[(size_t)b * QQ + q]    = (float)bidx;
        }
    }
}

extern "C" void kernel_launch(void* const* d_in, const int* in_sizes, int n_in,
                              void* d_out, int out_size, void* d_ws, size_t ws_size,
                              hipStream_t stream) {
    (void)in_sizes; (void)n_in; (void)out_size; (void)d_ws; (void)ws_size;
    const float4*  qry   = (const float4*)d_in[0];   // [8,16384,4] f32
    const float4*  tgt   = (const float4*)d_in[1];   // [8,256,4]   f32
    const uint8_t* masks = (const uint8_t*)d_in[2];  // [8,256,256,256] bool
    float*         out   = (float*)d_out;

    const dim3 grid(BB * (QQ / QPB));                // 2048 blocks
    box_matcher_kernel<<<grid, THREADS, 0, stream>>>(qry, tgt, masks, out);
}